// MeshLoss2D_28432683500146
// MI455X (gfx1250) — compile-verified
//
#include <hip/hip_runtime.h>

typedef __attribute__((ext_vector_type(2))) float v2f;
typedef __attribute__((ext_vector_type(8))) float v8f;

#define BB 4
#define MM 8192
#define NN 8192
#define SEGS 4
#define VT_PER_SEG ((NN / 16) / SEGS)   // 128 vertex tiles per segment

// Static device scratch (fully rewritten every call -> deterministic).
__device__ float g_va[BB * NN * 4];       // (vx, vy, vz, |v|^2) per vertex
__device__ float g_part[BB * SEGS * MM];  // per-segment partial min(|v|^2 - 2 p.v)

// v_min3_num_f32 without llvm.minnum's canonicalization self-max ops.
// Plain (non-volatile) asm: pure, lets the scheduler move it freely.
__device__ __forceinline__ float min3f(float a, float b, float c) {
  float r;
  asm("v_min3_num_f32 %0, %1, %2, %3" : "=v"(r) : "v"(a), "v"(b), "v"(c));
  return r;
}

// ---------------------------------------------------------------------------
// Kernel 0: build A-operand data: one float4 per vertex = [vx, vy, vz, |v|^2].
// Layout lets each lane load its WMMA A half-row (2 floats) with one b64 load.
// ---------------------------------------------------------------------------
__global__ __launch_bounds__(256) void prep_vertices_k(const float* __restrict__ vert) {
  int i = blockIdx.x * 256 + threadIdx.x;        // i = b*NN + n
  if (i >= BB * NN) return;
  int b = i / NN, n = i - b * NN;
  const float* vb = vert + (size_t)b * 3 * NN;
  float x = vb[n], y = vb[NN + n], z = vb[2 * NN + n];
  float4 o = make_float4(x, y, z, x * x + y * y + z * z);
  reinterpret_cast<float4*>(g_va)[i] = o;
}

// ---------------------------------------------------------------------------
// Kernel 1: per 16x16 tile, D = A x B  with A row = [vx,vy,vz,|v|^2] (rows =
// vertices) and B col = [-2px,-2py,-2pz,1] (cols = points), so
// D[n,m] = |v_n|^2 - 2 p_m . v_n.  min over vertices = min over D's 8 VGPRs
// (within lane) + one half-wave shuffle at the end.  |p|^2 added later.
//
// A-matrix 16x4 f32 layout: lanes 0-15 hold K=0,1; lanes 16-31 hold K=2,3
// (row M = lane%16).  B-matrix 4x16 mirrors this with N = lane%16.
// C/D layout: lane l covers column N=l%16, VGPR r covers row r (+8 if l>=16).
// ---------------------------------------------------------------------------
__global__ __launch_bounds__(256) void minsq_wmma_k(const float* __restrict__ pc) {
  const int lane = threadIdx.x & 31;
  const int wave = threadIdx.x >> 5;
  const int lp   = lane & 15;
  const bool hi  = lane >= 16;

  const int b     = blockIdx.x >> 6;   // 64 blocks per batch (SEGS*16)
  const int rem   = blockIdx.x & 63;
  const int seg   = rem >> 4;
  const int chunk = rem & 15;
  const int t0    = chunk * 32 + wave * 4;   // first of 4 point tiles for this wave

  const float* __restrict__ pcb = pc + (size_t)b * 3 * MM;
  const float* __restrict__ va  = g_va + (size_t)b * NN * 4;

  // Load 4 point tiles: build B operand registers once, keep resident.
  v2f   Bm[4];
  float acc[4];
#pragma unroll
  for (int p = 0; p < 4; ++p) {
    int m = (t0 + p) * 16 + lp;
    float px = pcb[m];
    float py = pcb[MM + m];
    float pz = pcb[2 * MM + m];
    Bm[p].x = hi ? (-2.0f * pz) : (-2.0f * px);   // K=2 : K=0
    Bm[p].y = hi ? 1.0f         : (-2.0f * py);   // K=3 : K=1
    acc[p]  = 3.402823466e38f;
  }

  const int vt0   = seg * VT_PER_SEG;
  const int vtEnd = vt0 + VT_PER_SEG;
  const int laneOff = lp * 4 + (hi ? 2 : 0);

  // 2-deep software pipeline: next A load in flight during WMMA + min fold.
  const float* ap0 = va + ((size_t)vt0 * 16 * 4 + laneOff);
  v2f A;
  A.x = ap0[0];
  A.y = ap0[1];

  for (int vt = vt0; vt < vtEnd; ++vt) {
    v2f Acur = A;
    int vtn = (vt + 1 < vtEnd) ? (vt + 1) : vt;   // branch-free clamp
    const float* apn = va + ((size_t)vtn * 16 * 4 + laneOff);
    A.x = apn[0];
    A.y = apn[1];

#pragma unroll
    for (int p = 0; p < 4; ++p) {
      v8f c = {0.f, 0.f, 0.f, 0.f, 0.f, 0.f, 0.f, 0.f};
      v8f d = __builtin_amdgcn_wmma_f32_16x16x4_f32(
          /*neg_a=*/false, Acur, /*neg_b=*/false, Bm[p],
          /*c_mod=*/(short)0, c, /*reuse_a=*/false, /*reuse_b=*/false);
      // fold 8 rows + running acc -> acc : exactly 4x v_min3_num_f32
      float x = min3f(d[0], d[1], d[2]);
      float y = min3f(d[3], d[4], d[5]);
      float z = min3f(d[6], d[7], acc[p]);
      acc[p]  = min3f(x, y, z);
    }
  }

  // Combine the two lane halves (rows 0-7 vs 8-15) and store partial min.
#pragma unroll
  for (int p = 0; p < 4; ++p) {
    float other = __shfl_xor(acc[p], 16, 32);
    float mn    = fminf(acc[p], other);
    if (!hi) {
      int m = (t0 + p) * 16 + lp;
      g_part[((size_t)b * SEGS + seg) * MM + m] = mn;
    }
  }
}

// ---------------------------------------------------------------------------
// Kernel 2: deterministic final reduction.  min over segments, add |p|^2,
// mask all-zero columns, masked mean per batch, mean over batches.
// ---------------------------------------------------------------------------
__global__ __launch_bounds__(1024) void final_reduce_k(const float* __restrict__ pc,
                                                       float* __restrict__ out) {
  __shared__ float sd[1024];
  __shared__ float sv[1024];
  const int tid = threadIdx.x;
  float total = 0.0f;
  for (int b = 0; b < BB; ++b) {
    float s = 0.0f, v = 0.0f;
    const float* pcb = pc + (size_t)b * 3 * MM;
    const float* pb  = g_part + (size_t)b * SEGS * MM;
    for (int m = tid; m < MM; m += 1024) {
      float mn = pb[m];
#pragma unroll
      for (int sgi = 1; sgi < SEGS; ++sgi) mn = fminf(mn, pb[(size_t)sgi * MM + m]);
      float px = pcb[m], py = pcb[MM + m], pz = pcb[2 * MM + m];
      bool valid = !(px == 0.0f && py == 0.0f && pz == 0.0f);
      float d2 = mn + (px * px + py * py + pz * pz);
      if (valid) { s += d2; v += 1.0f; }
    }
    sd[tid] = s;
    sv[tid] = v;
    __syncthreads();
    for (int off = 512; off > 0; off >>= 1) {
      if (tid < off) { sd[tid] += sd[tid + off]; sv[tid] += sv[tid + off]; }
      __syncthreads();
    }
    if (tid == 0) total += sd[0] / sv[0];
    __syncthreads();
  }
  if (tid == 0) out[0] = total * (1.0f / (float)BB);
}

extern "C" void kernel_launch(void* const* d_in, const int* in_sizes, int n_in,
                              void* d_out, int out_size, void* d_ws, size_t ws_size,
                              hipStream_t stream) {
  (void)in_sizes; (void)n_in; (void)out_size; (void)d_ws; (void)ws_size;
  const float* vert = (const float*)d_in[0];   // [B,3,128,64] f32
  const float* pc   = (const float*)d_in[1];   // [B,3,M] f32
  float* out        = (float*)d_out;           // scalar f32

  prep_vertices_k<<<(BB * NN + 255) / 256, 256, 0, stream>>>(vert);
  minsq_wmma_k<<<BB * SEGS * 16, 256, 0, stream>>>(pc);
  final_reduce_k<<<1, 1024, 0, stream>>>(pc, out);
}